// ICPMatcher_28484223107376
// MI455X (gfx1250) — compile-verified
//
#include <hip/hip_runtime.h>

typedef float v2f __attribute__((ext_vector_type(2)));
typedef float v8f __attribute__((ext_vector_type(8)));

#define WAVES_PER_BLOCK 8

// Pre-pass: build augmented matrices so one f32 WMMA computes the whole score.
//   augS[i] = [-2sx, -2sy, -2sz, 1]
//   augT[j] = [ tx,  ty,  tz, |t|^2 ]
//   score(i,j) = augS[i] . augT[j] = |t|^2 - 2 s.t   (distance minus |s|^2)
__global__ __launch_bounds__(256) void icp_prep(const float* __restrict__ src,
                                                const float* __restrict__ tgt,
                                                float4* __restrict__ augS,
                                                float4* __restrict__ augT,
                                                float* __restrict__ s2n,
                                                int n1, int n2) {
    int i = blockIdx.x * blockDim.x + threadIdx.x;
    if (i < n1) {
        float x = src[3 * i + 0], y = src[3 * i + 1], z = src[3 * i + 2];
        augS[i] = make_float4(-2.0f * x, -2.0f * y, -2.0f * z, 1.0f);
        s2n[i] = x * x + y * y + z * z;
    }
    if (i < n2) {
        float x = tgt[3 * i + 0], y = tgt[3 * i + 1], z = tgt[3 * i + 2];
        augT[i] = make_float4(x, y, z, x * x + y * y + z * z);
    }
}

// One block per 16-row source strip. 8 waves split the target tiles.
// Each tile: one v_wmma_f32_16x16x4_f32 produces the 16x16 score tile.
// C/D layout (f32): VGPR r, lanes 0-15 -> row M=r, lanes 16-31 -> row M=r+8,
// column N = lane & 15.
__global__ __launch_bounds__(256) void icp_nn(const float4* __restrict__ augS,
                                              const float4* __restrict__ augT,
                                              const float* __restrict__ s2n,
                                              float* __restrict__ outd,
                                              int* __restrict__ outi,
                                              int n2) {
    __shared__ float sv[WAVES_PER_BLOCK * 16];
    __shared__ int   si[WAVES_PER_BLOCK * 16];

    const int wave = threadIdx.x >> 5;
    const int lane = threadIdx.x & 31;
    const int half = lane >> 4;   // 0: K=0,1 half  1: K=2,3 half (A/B operand layout)
    const int lr   = lane & 15;
    const int row0 = blockIdx.x * 16;

    // A operand: 16x4 f32. Lane lr holds row M=lr; VGPR pair holds
    // (K=0,K=1) for lanes 0-15 and (K=2,K=3) for lanes 16-31.
    const float2* ap = (const float2*)(augS + row0 + lr);
    float2 af = ap[half];
    v2f A; A.x = af.x; A.y = af.y;

    const int ntiles = n2 >> 4;
    const int nk     = ntiles / WAVES_PER_BLOCK;   // wave-uniform trip count

    float bv[8];
    int   bi[8];
#pragma unroll
    for (int r = 0; r < 8; ++r) { bv[r] = INFINITY; bi[r] = 0; }

    // B operand pointer for tile t = wave: advances by 8 tiles (128 float4 =
    // 256 float2) per iteration; col advances by 8*16 = 128.
    const float2* bp = (const float2*)(augT + wave * 16 + lr) + half;
    int col = wave * 16 + lr;

#pragma unroll 2
    for (int k = 0; k < nk; ++k) {
        float2 bf = *bp;
        v2f B; B.x = bf.x; B.y = bf.y;

        v8f c = {0.f, 0.f, 0.f, 0.f, 0.f, 0.f, 0.f, 0.f};
        c = __builtin_amdgcn_wmma_f32_16x16x4_f32(
                /*neg_a=*/false, A, /*neg_b=*/false, B,
                /*c_mod=*/(short)0, c, /*reuse_a=*/false, /*reuse_b=*/false);

#pragma unroll
        for (int r = 0; r < 8; ++r) {
            float v  = c[r];
            bool  lt = v < bv[r];
            bv[r] = lt ? v   : bv[r];
            bi[r] = lt ? col : bi[r];
        }
        bp  += WAVES_PER_BLOCK * 16 * 2;   // float2 units
        col += WAVES_PER_BLOCK * 16;
    }

    // Tail for ntiles not divisible by WAVES_PER_BLOCK (not hit for n2=16384).
    {
        int t = nk * WAVES_PER_BLOCK + wave;
        if (t < ntiles) {
            const float2* tp = (const float2*)(augT + t * 16 + lr) + half;
            float2 bf = *tp;
            v2f B; B.x = bf.x; B.y = bf.y;
            v8f c = {0.f, 0.f, 0.f, 0.f, 0.f, 0.f, 0.f, 0.f};
            c = __builtin_amdgcn_wmma_f32_16x16x4_f32(
                    false, A, false, B, (short)0, c, false, false);
            int tc = t * 16 + lr;
#pragma unroll
            for (int r = 0; r < 8; ++r) {
                float v  = c[r];
                bool  lt = v < bv[r];
                bv[r] = lt ? v  : bv[r];
                bi[r] = lt ? tc : bi[r];
            }
        }
    }

    // Columns of each row live across the 16 lanes of a half-wave:
    // branchless xor-reduce (min, tie -> lowest index) within each 16-lane group.
#pragma unroll
    for (int r = 0; r < 8; ++r) {
        float v = bv[r];
        int   i = bi[r];
#pragma unroll
        for (int m = 8; m >= 1; m >>= 1) {
            float ov = __shfl_xor(v, m, 32);
            int   oi = __shfl_xor(i, m, 32);
            bool take = (ov < v) | ((ov == v) & (oi < i));
            v = take ? ov : v;
            i = take ? oi : i;
        }
        if (lr == 0) {
            int row = r + 8 * half;   // row within this 16-row strip
            sv[wave * 16 + row] = v;
            si[wave * 16 + row] = i;
        }
    }
    __syncthreads();

    // Combine the 8 waves' partial results, branchless, tie -> lowest index.
    if (threadIdx.x < 16) {
        int row = threadIdx.x;
        float v = sv[row];
        int   i = si[row];
#pragma unroll
        for (int w = 1; w < WAVES_PER_BLOCK; ++w) {
            float ov = sv[w * 16 + row];
            int   oi = si[w * 16 + row];
            bool take = (ov < v) | ((ov == v) & (oi < i));
            v = take ? ov : v;
            i = take ? oi : i;
        }
        int g = row0 + row;
        outd[g] = v + s2n[g];   // restore |s|^2 to get the true min distance
        outi[g] = i;
    }
}

extern "C" void kernel_launch(void* const* d_in, const int* in_sizes, int n_in,
                              void* d_out, int out_size, void* d_ws, size_t ws_size,
                              hipStream_t stream) {
    (void)n_in; (void)out_size; (void)ws_size;
    const float* src = (const float*)d_in[0];
    const float* tgt = (const float*)d_in[1];
    const int n1 = in_sizes[0] / 3;   // 16384
    const int n2 = in_sizes[1] / 3;   // 16384

    float4* augS = (float4*)d_ws;
    float4* augT = augS + n1;
    float*  s2n  = (float*)(augT + n2);

    float* outd = (float*)d_out;
    int*   outi = (int*)(outd + n1);

    const int n = n1 > n2 ? n1 : n2;
    icp_prep<<<(n + 255) / 256, 256, 0, stream>>>(src, tgt, augS, augT, s2n, n1, n2);
    icp_nn<<<n1 / 16, 256, 0, stream>>>(augS, augT, s2n, outd, outi, n2);
}